// CTCDecoder_71820443124341
// MI455X (gfx1250) — compile-verified
//
#include <hip/hip_runtime.h>
#include <hip/hip_bf16.h>

// CTC loss for MI455X (gfx1250, wave32, WMMA).
//
// Roofline: GEMM = 41 GFLOP fp32 dominates; bytes needed are only ~35 MB if we
// never materialize logits[B,T,V] (320 MB). We fuse GEMM + streaming logsumexp
// using V_WMMA_F32_16X16X4_F32 (exact fp32), gather extended-label logits with
// a second WMMA kernel, then run the serial-in-T CTC scan with one workgroup
// per batch element (alpha in LDS, 1 barrier/step).

typedef __attribute__((ext_vector_type(2))) float v2f;
typedef __attribute__((ext_vector_type(8))) float v8f;

#define NEGF (-1e30f)

constexpr int B_ = 32;
constexpr int T_ = 500;
constexpr int D_ = 256;   // K dim; 256/4 = 64 WMMA k-steps
constexpr int V_ = 5000;
constexpr int L_ = 100;
constexpr int S_ = 2 * L_ + 1;   // 201 extended states
constexpr int SP = 208;          // padded S stride (13 tiles of 16)
constexpr int TTILES = (T_ + 15) / 16;   // 32
constexpr int STILES = SP / 16;          // 13
constexpr int VTILES = (V_ + 15) / 16;   // 313

__device__ __forceinline__ float laddexp(float a, float b) {
    float m = fmaxf(a, b);
    float d = fabsf(a - b);
    return m + log1pf(__expf(-d));
}

// ---------------------------------------------------------------------------
// Kernel A: fused  logits-tile = x_tile @ W_tile^T  (WMMA f32 16x16x4)
//           + streaming logsumexp over V.  Grid: (t-tile, b). 8 waves/WG,
//           each wave strides over the 313 v-tiles.
// ---------------------------------------------------------------------------
__global__ void __launch_bounds__(256)
gemm_lse_kernel(const float* __restrict__ x, const float* __restrict__ W,
                const float* __restrict__ bias, float* __restrict__ lse)
{
    const int ttile = blockIdx.x;
    const int b     = blockIdx.y;
    const int tid   = threadIdx.x;
    const int wave  = tid >> 5;
    const int lane  = tid & 31;
    const int trow  = lane & 15;           // M index (time row)
    const int kb    = (lane >> 4) * 2;     // K sub-offset per ISA A/B layout
    const int t0    = ttile * 16;
    const int tIdx  = min(t0 + trow, T_ - 1);   // clamp tail tile

    // Preload the whole A tile (16 x 256 fp32) into registers: 64 x v2f.
    const float* xrow = x + ((size_t)b * T_ + tIdx) * D_ + kb;
    v2f a[64];
#pragma unroll
    for (int kk = 0; kk < 64; ++kk)
        a[kk] = *(const v2f*)(xrow + kk * 4);

    // Per-lane streaming logsumexp state for this lane's 8 M-rows.
    float mx[8], sm[8];
#pragma unroll
    for (int j = 0; j < 8; ++j) { mx[j] = NEGF; sm[j] = 0.0f; }

    for (int vt = wave; vt < VTILES; vt += 8) {
        const int v0   = vt * 16;
        const int vcol = v0 + (lane & 15);               // N index (vocab col)
        const int vIdx = min(vcol, V_ - 1);
        const float* wrow = W + (size_t)vIdx * D_ + kb;

        v8f acc = {};
#pragma unroll
        for (int kk = 0; kk < 64; ++kk) {
            v2f bv = *(const v2f*)(wrow + kk * 4);
            acc = __builtin_amdgcn_wmma_f32_16x16x4_f32(
                false, a[kk], false, bv, (short)0, acc, false, false);
        }

        const bool  valid = vcol < V_;
        const float bb    = bias[vIdx];
#pragma unroll
        for (int j = 0; j < 8; ++j) {
            float l  = valid ? (acc[j] + bb) : NEGF;
            float nm = fmaxf(mx[j], l);
            sm[j] = sm[j] * __expf(mx[j] - nm) + __expf(l - nm);
            mx[j] = nm;
        }
    }

    // Cross-lane / cross-wave combine of (max, sumexp) per time row.
    __shared__ float red[2][8][32][8];
#pragma unroll
    for (int j = 0; j < 8; ++j) {
        red[0][wave][lane][j] = mx[j];
        red[1][wave][lane][j] = sm[j];
    }
    __syncthreads();

    if (tid < 16) {
        const int m  = tid;               // time row within tile
        const int jj = m & 7;
        const int lo = (m >= 8) ? 16 : 0; // which half-wave owns this row
        float M = NEGF, S = 0.0f;
        for (int w = 0; w < 8; ++w)
            for (int l = 0; l < 16; ++l) {
                float M2 = red[0][w][lo + l][jj];
                float S2 = red[1][w][lo + l][jj];
                float nm = fmaxf(M, M2);
                S = S * __expf(M - nm) + S2 * __expf(M2 - nm);
                M = nm;
            }
        const int t = t0 + m;
        if (t < T_) lse[b * T_ + t] = M + __logf(S);
    }
}

// ---------------------------------------------------------------------------
// Kernel B: extended-label logits via WMMA with gathered W rows.
//   lab[t][b][s] = dot(x[b,t], W[ext[b,s]]) + bias[ext] - lse[b,t]
// Grid: (b, s-tile, t-group of 8 waves). ext: s odd -> labels[(s-1)/2], else 0.
// ---------------------------------------------------------------------------
__global__ void __launch_bounds__(256)
label_logits_kernel(const float* __restrict__ x, const float* __restrict__ W,
                    const float* __restrict__ bias, const int* __restrict__ labels,
                    const float* __restrict__ lse, float* __restrict__ lab)
{
    const int b     = blockIdx.x;
    const int stile = blockIdx.y;   // 0..12
    const int tgrp  = blockIdx.z;   // 0..3
    const int tid   = threadIdx.x;
    const int wave  = tid >> 5;
    const int lane  = tid & 31;
    const int ttile = tgrp * 8 + wave;   // 0..31
    const int t0    = ttile * 16;
    const int trow  = lane & 15;
    const int kb    = (lane >> 4) * 2;
    const int tIdx  = min(t0 + trow, T_ - 1);

    const float* xrow = x + ((size_t)b * T_ + tIdx) * D_ + kb;
    v2f a[64];
#pragma unroll
    for (int kk = 0; kk < 64; ++kk)
        a[kk] = *(const v2f*)(xrow + kk * 4);

    const int scol = stile * 16 + (lane & 15);
    int vext = 0;                                  // blank
    if ((scol & 1) && scol < S_) {
        vext = labels[b * L_ + (scol >> 1)];
        vext = min(max(vext, 0), V_ - 1);
    }
    const float* wrow = W + (size_t)vext * D_ + kb;

    v8f acc = {};
#pragma unroll
    for (int kk = 0; kk < 64; ++kk) {
        v2f bv = *(const v2f*)(wrow + kk * 4);
        acc = __builtin_amdgcn_wmma_f32_16x16x4_f32(
            false, a[kk], false, bv, (short)0, acc, false, false);
    }

    const float bb = bias[vext];
#pragma unroll
    for (int j = 0; j < 8; ++j) {
        const int m = j + ((lane >= 16) ? 8 : 0);
        const int t = t0 + m;
        if (t < T_)
            lab[((size_t)t * B_ + b) * SP + scol] = acc[j] + bb - lse[b * T_ + t];
    }
}

// ---------------------------------------------------------------------------
// Kernel C: CTC forward recursion. One workgroup per batch element; thread s
// owns extended state s; alpha double-buffered in LDS, 1 barrier per step.
// ---------------------------------------------------------------------------
__global__ void __launch_bounds__(224)
ctc_scan_kernel(const float* __restrict__ lab, const int* __restrict__ in_lens,
                const int* __restrict__ lab_lens, const int* __restrict__ labels,
                float* __restrict__ ll)
{
    const int b = blockIdx.x;
    const int s = threadIdx.x;    // 0..223 (201 active)
    __shared__ float alp[2][224];

    int  ext_s  = 0;
    bool allow2 = false;
    if ((s & 1) && s < S_) {
        ext_s = labels[b * L_ + (s >> 1)];
        int em2 = (s >= 3) ? labels[b * L_ + (s >> 1) - 1] : -1;
        allow2 = (ext_s != em2);   // ext_s != blank guaranteed (s odd)
    }
    const int myLen = in_lens[b];
    const float* labb = lab + (size_t)b * SP;   // + t*B_*SP + s per step

    float a0 = NEGF;
    if (s == 0) a0 = labb[0];
    else if (s == 1) a0 = labb[1];
    alp[0][s] = a0;
    __syncthreads();

    int p = 0;
    for (int t = 1; t < T_; ++t) {
        float cur = alp[p][s];
        float nv  = cur;                       // frozen past input length
        if (t < myLen && s < S_) {
            float a1 = (s >= 1) ? alp[p][s - 1] : NEGF;
            float aa = laddexp(cur, a1);
            if (allow2) {
                float a2 = (s >= 2) ? alp[p][s - 2] : NEGF;
                aa = laddexp(aa, a2);
            }
            nv = aa + labb[(size_t)t * B_ * SP + s];
        }
        if (s >= S_) nv = NEGF;
        alp[1 - p][s] = nv;
        __syncthreads();
        p ^= 1;
    }

    if (s == 0) {
        const int len = lab_lens[b];
        float e1 = alp[p][2 * len];
        float e2 = alp[p][2 * len - 1];
        ll[b] = laddexp(e1, e2);
    }
}

// ---------------------------------------------------------------------------
// Kernel D: out = -sum_b ll[b] / L
// ---------------------------------------------------------------------------
__global__ void finalize_kernel(const float* __restrict__ ll, float* __restrict__ out)
{
    if (threadIdx.x == 0) {
        float sum = 0.0f;
        for (int i = 0; i < B_; ++i) sum += ll[i];
        out[0] = -sum / (float)L_;
    }
}

// ---------------------------------------------------------------------------
extern "C" void kernel_launch(void* const* d_in, const int* in_sizes, int n_in,
                              void* d_out, int out_size, void* d_ws, size_t ws_size,
                              hipStream_t stream)
{
    const float* x        = (const float*)d_in[0];   // [B,T,D]
    const int*   in_lens  = (const int*)  d_in[1];   // [B]
    const int*   labels   = (const int*)  d_in[2];   // [B,L]
    const int*   lab_lens = (const int*)  d_in[3];   // [B]
    const float* W        = (const float*)d_in[4];   // [V,D]
    const float* bias     = (const float*)d_in[5];   // [V]
    float*       out      = (float*)d_out;

    // Workspace layout (all fully written before use):
    //   lse : B*T floats   @ 0           (64 KB)
    //   ll  : B   floats   @ 64 KiB
    //   lab : T*B*SP floats@ 64 KiB + 1 KiB  (~12.7 MB)
    char*  ws  = (char*)d_ws;
    float* lse = (float*)(ws);
    float* ll  = (float*)(ws + (64u << 10));
    float* lab = (float*)(ws + (64u << 10) + 1024u);

    gemm_lse_kernel<<<dim3(TTILES, B_), 256, 0, stream>>>(x, W, bias, lse);
    label_logits_kernel<<<dim3(B_, STILES, 4), 256, 0, stream>>>(x, W, bias, labels, lse, lab);
    ctc_scan_kernel<<<dim3(B_), 224, 0, stream>>>(lab, in_lens, lab_lens, labels, ll);
    finalize_kernel<<<dim3(1), 32, 0, stream>>>(ll, out);
}